// Attention_8014408974955
// MI455X (gfx1250) — compile-verified
//
#include <hip/hip_runtime.h>
#include <math.h>

// CDNA5 / gfx1250: fp32 WMMA 16x16x4. A = 16x4 (2 VGPRs), B = 4x16 (2 VGPRs),
// C/D = 16x16 f32 (8 VGPRs).
typedef __attribute__((ext_vector_type(2))) float v2f;
typedef __attribute__((ext_vector_type(8))) float v8f;

#define HIDDEN   1024
#define N_INTER  256
#define WINDOW   64
// STD_SQUARED = (64/2)^2 = 1024 -> 2*STD_SQUARED = 2048
#define TWO_STD2 2048.0f

// ---------------------------------------------------------------------------
// Kernel 1: fc1 GEMV, distributed over 16 workgroups (one WGP each).
//   Block b computes p[16b .. 16b+15] = tanh(h_t . fc1_w[row]^T + fc1_b[row]).
//   8 waves split K=1024 into 128-element chunks; each wave runs 32 steps of
//   V_WMMA_F32_16X16X4_F32 with A-row-0 = h_t[k..k+3], B = fc1_w^T chunk.
//   Per-wave partials (D row 0) are reduced through LDS, then tanh+bias.
//   Spreads the 1 MB fc1_w stream across 16 WGPs instead of one.
// ---------------------------------------------------------------------------
__global__ __launch_bounds__(256) void attn_fc1_kernel(
    const float* __restrict__ h_t,
    const float* __restrict__ fc1_w,
    const float* __restrict__ fc1_b,
    float*       __restrict__ p_out)     // ws: 256 floats
{
    __shared__ float part[8][16];

    const int tid   = threadIdx.x;
    const int lane  = tid & 31;
    const int wave  = tid >> 5;           // 0..7
    const int row0  = blockIdx.x * 16;    // first fc1 output row of this block
    const int m     = lane & 15;          // N index (output column in D)
    const int sub   = (lane >> 4) << 1;   // K sub-offset: 0 (lanes 0-15) / 2 (16-31)
    const int kbase = wave * (HIDDEN / 8);

    v8f c = {0.f, 0.f, 0.f, 0.f, 0.f, 0.f, 0.f, 0.f};
    const float* __restrict__ wrow = fc1_w + (size_t)(row0 + m) * HIDDEN;

    for (int kk = 0; kk < HIDDEN / 8; kk += 4) {
        const int k = kbase + kk;
        // A (16x4): row M=0 holds h_t[k..k+3]; rows 1..15 are zero.
        float hx = h_t[k + sub];
        float hy = h_t[k + sub + 1];
        v2f a;
        a.x = (m == 0) ? hx : 0.0f;
        a.y = (m == 0) ? hy : 0.0f;
        // B (4x16): B[kk][n] = fc1_w[row0+n][k+kk]  (fc1_w^T chunk)
        v2f b;
        b.x = wrow[k + sub];
        b.y = wrow[k + sub + 1];
        c = __builtin_amdgcn_wmma_f32_16x16x4_f32(false, a, false, b,
                                                  (short)0, c, false, false);
    }

    // D row 0 lives in accumulator element 0 of lanes 0..15 (N = lane).
    if (lane < 16) part[wave][lane] = c[0];
    __syncthreads();

    if (tid < 16) {
        float s = 0.0f;
        #pragma unroll
        for (int w = 0; w < 8; ++w) s += part[w][tid];
        p_out[row0 + tid] = tanhf(s + fc1_b[row0 + tid]);
    }
}

// ---------------------------------------------------------------------------
// Kernel 2: fc2 + sigmoid + window params (redundantly per block, trivial),
//   then c_t[col] = e_t * sum_{r=w_start..w_end} h_s[r][col].
//   8 blocks x 8 waves = 64 waves, one 16-column tile per wave.
//   WMMA with A-row-0 = [e_t,e_t,e_t,e_t] (edge-masked), B = 4x16 h_s tile.
//   (softmax over the singleton axis in the reference == all-ones weights,
//    so the 256 MB score GEMV is dead code and skipped entirely.)
// ---------------------------------------------------------------------------
__global__ __launch_bounds__(256) void attn_ctx_kernel(
    const float* __restrict__ h_s,
    const float* __restrict__ fc2_w,
    const float* __restrict__ fc2_b,
    const int*   __restrict__ S_ptr,
    const float* __restrict__ p_in,      // ws: 256 floats from kernel 1
    float*       __restrict__ out)
{
    __shared__ int   s_wst, s_wen;
    __shared__ float s_et;

    const int tid  = threadIdx.x;
    const int lane = tid & 31;
    const int wave = tid >> 5;                 // 0..7
    const int tile = blockIdx.x * 8 + wave;    // 0..63
    const int col0 = tile * 16;
    const int m    = lane & 15;
    const int sub  = (lane >> 4) << 1;

    // Wave 0 of each block computes the scalar head (cheap, all L2 hits).
    if (tid < 32) {
        float partial = 0.0f;
        #pragma unroll
        for (int j = 0; j < N_INTER / 32; ++j) {
            int i = tid + j * 32;
            partial += p_in[i] * fc2_w[i];
        }
        #pragma unroll
        for (int off = 16; off > 0; off >>= 1)
            partial += __shfl_xor(partial, off, 32);

        if (tid == 0) {
            const int S = S_ptr[0];
            float logit = partial + fc2_b[0];
            float pt = (float)S * (1.0f / (1.0f + expf(-logit)));
            int wst = (int)ceilf(pt - (float)WINDOW);
            if (wst < 0) wst = 0;
            int wen = (int)floorf(pt + (float)WINDOW);
            if (wen > S - 1) wen = S - 1;
            s_wst = wst;
            s_wen = wen;
            s_et  = expf(((float)S - pt) / TWO_STD2);
        }
    }
    __syncthreads();

    // Wave-uniform window parameters (keep EXEC all-ones through the loop).
    const int   wst = __builtin_amdgcn_readfirstlane(s_wst);
    const int   wen = __builtin_amdgcn_readfirstlane(s_wen);
    const float et  = __int_as_float(
        __builtin_amdgcn_readfirstlane(__float_as_int(s_et)));
    const int n = wen - wst + 1;               // 1..129 rows in window

    v8f c = {0.f, 0.f, 0.f, 0.f, 0.f, 0.f, 0.f, 0.f};

    for (int k = 0; k < n; k += 4) {
        // A row 0: e_t where the row is inside the window, else 0.
        v2f a;
        a.x = ((m == 0) && (k + sub)     < n) ? et : 0.0f;
        a.y = ((m == 0) && (k + sub + 1) < n) ? et : 0.0f;
        // B: 4 rows of h_s at columns [col0, col0+16). Clamp OOB rows to wen
        // (their A weight is 0, so the clamped value contributes nothing).
        int r0 = wst + k + sub;
        int r1 = r0 + 1;
        if (r0 > wen) r0 = wen;
        if (r1 > wen) r1 = wen;
        v2f b;
        b.x = h_s[(size_t)r0 * HIDDEN + col0 + m];
        b.y = h_s[(size_t)r1 * HIDDEN + col0 + m];
        c = __builtin_amdgcn_wmma_f32_16x16x4_f32(false, a, false, b,
                                                  (short)0, c, false, false);
    }

    if (lane < 16) out[col0 + lane] = c[0];
}

// ---------------------------------------------------------------------------
// Host-side launcher.
// Inputs (setup_inputs order):
//   0: source_hiddens [65536,1024] f32   1: target_hidden [1,1024] f32
//   2: fc1_w [256,1024] f32              3: fc1_b [256] f32
//   4: fc2_w [1,256] f32                 5: fc2_b [1] f32
//   6: source_sentence_length (int, 1 elem)
// Output: c_t [1,1024] f32
// Workspace: p[256] floats (1 KB).
// ---------------------------------------------------------------------------
extern "C" void kernel_launch(void* const* d_in, const int* in_sizes, int n_in,
                              void* d_out, int out_size, void* d_ws, size_t ws_size,
                              hipStream_t stream) {
    (void)in_sizes; (void)n_in; (void)out_size; (void)ws_size;
    const float* h_s   = (const float*)d_in[0];
    const float* h_t   = (const float*)d_in[1];
    const float* fc1_w = (const float*)d_in[2];
    const float* fc1_b = (const float*)d_in[3];
    const float* fc2_w = (const float*)d_in[4];
    const float* fc2_b = (const float*)d_in[5];
    const int*   S_ptr = (const int*)d_in[6];
    float* out = (float*)d_out;
    float* p   = (float*)d_ws;

    attn_fc1_kernel<<<16, 256, 0, stream>>>(h_t, fc1_w, fc1_b, p);
    attn_ctx_kernel<<<8, 256, 0, stream>>>(h_s, fc2_w, fc2_b, S_ptr, p, out);
}